// ProbabilisticPatching_27152783245864
// MI455X (gfx1250) — compile-verified
//
#include <hip/hip_runtime.h>
#include <stdint.h>

// ProbabilisticPatching for MI455X (gfx1250, wave32).
//   out[b][p][0:512]   = x[b] * mask(b,p)        (f32)
//   out[b][p][512:1024]= mask(b,p)               (f32), |mask| = 32 (top-k of logp+gumbel)
//
// CDNA5-specific paths (all verified in the round-1 histogram):
//   * TDM tensor_load_to_lds: DMA the 64x512 f32 log-softmax table (128 KiB)
//     into LDS once per workgroup; s_wait_tensorcnt to sync.
//   * wave32 ballot-based radix select for exact top-32 per row
//     (v_cmp -> SGPR mask, s_bcnt/s_add co-issue on the SALU).
//   * non-temporal global_store_b128 for the 512 MiB streaming output
//     (store-bandwidth floor ~23 us at 23.3 TB/s).
// New this round: early-exit bit descent — stop as soon as a prefix separates
// exactly 32 keys (common case, no ties), halving selection VALU cost.

#define BATCH  2048
#define NPATCH 64
#define NFEAT  512
#define PLEN   32

typedef __attribute__((ext_vector_type(4))) uint32_t u32x4;
typedef __attribute__((ext_vector_type(8))) uint32_t u32x8;
typedef __attribute__((ext_vector_type(4))) float    f32x4;

__device__ __forceinline__ uint32_t mix3(uint32_t a, uint32_t b, uint32_t c) {
  uint32_t h = a * 0x9E3779B1u ^ b * 0x85EBCA77u ^ c * 0xC2B2AE3Du;
  h ^= h >> 15; h *= 0x2C1B3C6Du;
  h ^= h >> 12; h *= 0x297A2D39u;
  h ^= h >> 15;
  return h;
}

// ---------------- Kernel 1: logp = log_softmax(weights, axis=-1) -------------
__global__ __launch_bounds__(256) void pp_logsoftmax(const float* __restrict__ w,
                                                     float* __restrict__ logp) {
  __shared__ float red[256];
  const int p = blockIdx.x;     // 0..63
  const int t = threadIdx.x;    // 0..255, each thread handles 2 features
  const float a = w[p * NFEAT + t];
  const float b = w[p * NFEAT + 256 + t];

  red[t] = fmaxf(a, b);
  __syncthreads();
  for (int s = 128; s > 0; s >>= 1) {
    if (t < s) red[t] = fmaxf(red[t], red[t + s]);
    __syncthreads();
  }
  const float rowmax = red[0];
  __syncthreads();

  red[t] = __expf(a - rowmax) + __expf(b - rowmax);
  __syncthreads();
  for (int s = 128; s > 0; s >>= 1) {
    if (t < s) red[t] += red[t + s];
    __syncthreads();
  }
  const float lse = __logf(red[0]) + rowmax;

  logp[p * NFEAT + t]       = a - lse;
  logp[p * NFEAT + 256 + t] = b - lse;
}

// ---------------- Kernel 2: gumbel top-k + patch emit ------------------------
// One workgroup (8 waves) per batch row b. Wave w handles patches p = 8w..8w+7.
// Lane L owns features f = ch*128 + L*4 + c (ch,c in 0..3) -> contiguous B128 IO.
__global__ __launch_bounds__(256) void pp_patch_topk(const float* __restrict__ x,
                                                     const float* __restrict__ logp,
                                                     const int* __restrict__ seed_p,
                                                     float* __restrict__ out) {
  extern __shared__ float s_logp[];  // NPATCH*NFEAT floats = 128 KiB (of 320 KiB/WGP)
  const int b    = (int)blockIdx.x;
  const int tid  = (int)threadIdx.x;
  const int wave = tid >> 5;
  const int lane = tid & 31;

  // --- TDM: DMA logp[64x512] (global, L2-hot) -> LDS. Wave 0 issues; EXEC
  // is ignored by tensor ops, so a uniform branch is sufficient.
  if (wave == 0) {
    const uint32_t lds_off = (uint32_t)(uintptr_t)(void*)s_logp; // LDS byte offset
    const uint64_t ga      = (uint64_t)(uintptr_t)logp;
    u32x4 g0;
    g0[0] = 1u;                                            // count=1, user descriptor
    g0[1] = lds_off;                                       // lds_addr
    g0[2] = (uint32_t)ga;                                  // global_addr[31:0]
    g0[3] = ((uint32_t)(ga >> 32) & 0x01FFFFFFu)           // global_addr[56:32]
            | 0x80000000u;                                 // type=2 ("image")
    u32x8 g1;
    g1[0] = 0x00020000u;                   // wg_mask=0 (not in cluster), data_size=4B
    g1[1] = (uint32_t)NFEAT << 16;         // tensor_dim0[15:0] in bits[31:16]
    g1[2] = (uint32_t)NPATCH << 16;        // tensor_dim0 hi=0 | tensor_dim1=64
    g1[3] = (uint32_t)NFEAT << 16;         // tensor_dim1 hi=0 | tile_dim0=512
    g1[4] = (uint32_t)NPATCH;              // tile_dim1=64, tile_dim2=0
    g1[5] = (uint32_t)NFEAT;               // tensor_dim0_stride lo32 = 512
    g1[6] = 0u;                            // stride hi, tensor_dim1_stride lo
    g1[7] = 0u;
    asm volatile("tensor_load_to_lds %0, %1" :: "s"(g0), "s"(g1) : "memory");
  }

  // Overlap with the DMA: x[b] row -> registers (same 16 slots for all 64 p).
  const uint32_t seed = (uint32_t)seed_p[0];
  f32x4 xr[4];
  {
    const f32x4* x4 = (const f32x4*)(x + (size_t)b * NFEAT);
#pragma unroll
    for (int ch = 0; ch < 4; ++ch) xr[ch] = x4[ch * 32 + lane];
  }

  if (wave == 0) __builtin_amdgcn_s_wait_tensorcnt(0);  // TENSORcnt is per-wave
  __syncthreads();                                      // publish LDS to all waves

  f32x4* out4 = (f32x4*)out;

  for (int pi = 0; pi < 8; ++pi) {
    const int p = wave * 8 + pi;
    const uint32_t row = (uint32_t)(b * NPATCH + p);

    // --- scores -> order-preserving u32 keys (16 per lane) ---
    uint32_t key[16];
    const f32x4* lp4 = (const f32x4*)(s_logp + p * NFEAT);
#pragma unroll
    for (int ch = 0; ch < 4; ++ch) {
      const f32x4 lp = lp4[ch * 32 + lane];               // ds_load_b128
#pragma unroll
      for (int c = 0; c < 4; ++c) {
        const uint32_t f = (uint32_t)(ch * 128 + lane * 4 + c);
        const uint32_t h = mix3(row, f, seed);
        const float u = fmaf((float)(h >> 8), 0x1p-24f, 0x1p-25f); // in (0,1)
        const float g = -__logf(-__logf(u));                       // gumbel
        const float s = lp[c] + g;
        uint32_t kb = __float_as_uint(s);
        kb = (kb & 0x80000000u) ? ~kb : (kb | 0x80000000u);        // monotone map
        key[ch * 4 + c] = kb;
      }
    }

    // --- radix bit-descent with early exit ---
    // Invariant: count(key >= T) >= 32. Stop as soon as the count is exactly
    // 32 (prefix already separates the top-32); otherwise full descent pins
    // T = exact 32nd-largest key and ties are trimmed below.
    uint32_t T = 0u;
    int cntT = NFEAT;                      // count(key >= T), uniform
    for (int bit = 31; bit >= 0; --bit) {
      if (cntT == PLEN) break;             // uniform scalar branch
      const uint32_t cand = T | (1u << bit);
      int cnt = 0;
#pragma unroll
      for (int j = 0; j < 16; ++j)
        cnt += __builtin_popcount(__builtin_amdgcn_ballot_w32(key[j] >= cand));
      if (cnt >= PLEN) { T = cand; cntT = cnt; }
    }

    uint32_t selbits = 0u;
    if (cntT == PLEN) {
      // Fast path: threshold separates exactly 32 keys; pure per-lane compares.
#pragma unroll
      for (int j = 0; j < 16; ++j)
        if (key[j] >= T) selbits |= 1u << j;
    } else {
      // Tie path: T is the exact 32nd-largest key; take all >T plus the first
      // (32 - count(>T)) ties in (j, lane) order via ballot prefix counts.
      int ngt = 0;
#pragma unroll
      for (int j = 0; j < 16; ++j) {
        const bool gt = key[j] > T;
        if (gt) selbits |= 1u << j;
        ngt += __builtin_popcount(__builtin_amdgcn_ballot_w32(gt));
      }
      const int need = PLEN - ngt;         // uniform, in (0, 32]
      int acc = 0;
#pragma unroll
      for (int j = 0; j < 16; ++j) {
        const uint32_t m = __builtin_amdgcn_ballot_w32(key[j] == T);
        const int before = __builtin_popcount(m & ((1u << lane) - 1u));
        if (((m >> lane) & 1u) && (acc + before) < need) selbits |= 1u << j;
        acc += __builtin_popcount(m);
      }
    }

    // --- emit [patches | mask] with non-temporal B128 stores ---
    const size_t rowbase4 = (size_t)row * 256;   // 1024 f32 = 256 f32x4 per row
#pragma unroll
    for (int ch = 0; ch < 4; ++ch) {
      f32x4 pv, mv;
#pragma unroll
      for (int c = 0; c < 4; ++c) {
        const float msk = ((selbits >> (ch * 4 + c)) & 1u) ? 1.0f : 0.0f;
        mv[c] = msk;
        pv[c] = xr[ch][c] * msk;
      }
      __builtin_nontemporal_store(pv, out4 + rowbase4 + ch * 32 + lane);        // patches
      __builtin_nontemporal_store(mv, out4 + rowbase4 + 128 + ch * 32 + lane);  // mask
    }
  }
}

// ---------------- launch -----------------------------------------------------
extern "C" void kernel_launch(void* const* d_in, const int* in_sizes, int n_in,
                              void* d_out, int out_size, void* d_ws, size_t ws_size,
                              hipStream_t stream) {
  (void)in_sizes; (void)n_in; (void)out_size; (void)ws_size;
  const float* x      = (const float*)d_in[0];   // [2048, 512] f32
  const float* w      = (const float*)d_in[1];   // [64, 512]   f32
  const int*   seed   = (const int*)d_in[2];     // scalar
  float*       out    = (float*)d_out;           // [2048, 64, 1024] f32
  float*       logp   = (float*)d_ws;            // 128 KiB scratch

  pp_logsoftmax<<<NPATCH, 256, 0, stream>>>(w, logp);
  pp_patch_topk<<<BATCH, 256, NPATCH * NFEAT * sizeof(float), stream>>>(x, logp, seed, out);
}